// CannyEdge_84739704750346
// MI455X (gfx1250) — compile-verified
//
#include <hip/hip_runtime.h>
#include <math.h>
#include <stdint.h>

// ---------------------------------------------------------------------------
// Canny edge detection, fused pipeline for MI455X (gfx1250, wave32).
//
// Pass 1: separable 5-tap Gaussian as banded-matrix GEMMs on 16x16 tiles
//         via V_WMMA_F32_16X16X4_F32 (K=20 -> 5 chained K=4 WMMAs).
//         Interior tiles are fetched with the Tensor Data Mover
//         (tensor_load_to_lds + s_wait_tensorcnt); boundary tiles use
//         manual zero-padded loads.
// Pass 2: Sobel + |grad| + direction index + per-image max (atomicMax on bits)
// Pass 3: NMS + per-image normalize + global thin-max
// Pass 4: double threshold -> uint8 state {0,1=weak,2=strong}
// Pass 5: x24 hysteresis sweeps, intra-tile convergence in LDS (monotone,
//         in-place on uint8 state -> only 32MB traffic per sweep)
// Pass 6: finalize to float {255, 0}
// ---------------------------------------------------------------------------

typedef float v2f __attribute__((ext_vector_type(2)));
typedef float v8f __attribute__((ext_vector_type(8)));
typedef unsigned int u32x4 __attribute__((ext_vector_type(4)));
typedef int i32x4 __attribute__((ext_vector_type(4)));
typedef int i32x8 __attribute__((ext_vector_type(8)));

#define IMG_H 1024
#define IMG_W 1024
#define BATCH 16
#define NPIX  ((size_t)BATCH * IMG_H * IMG_W)
#define HW    (IMG_H * IMG_W)   // 1<<20

#define LOW_T   0.00392f
#define HIGH_T  0.15f
#define HYST_ITERS 24

#if defined(__has_builtin)
#if __has_builtin(__builtin_amdgcn_tensor_load_to_lds) && \
    __has_builtin(__builtin_amdgcn_s_wait_tensorcnt)
#define HAVE_TDM 1
#endif
#endif
#ifndef HAVE_TDM
#define HAVE_TDM 0
#endif

static __device__ __forceinline__ v8f wmma4(v2f a, v2f b, v8f c) {
  // D = A(16x4) * B(4x16) + C(16x16), fp32
  return __builtin_amdgcn_wmma_f32_16x16x4_f32(
      /*neg_a=*/false, a, /*neg_b=*/false, b,
      /*c_mod=*/(short)0, c, /*reuse_a=*/false, /*reuse_b=*/false);
}

// Unnormalized gaussian window exp(-0.5*((i-2)^2)), i in 0..4; 0 outside band.
// Branchless: must lower to v_cmp + v_cndmask, NOT exec-mask branches.
static __device__ __forceinline__ float gband(int d) {
  float r = (d == 2) ? 1.0f : 0.0f;
  r = ((d == 1) | (d == 3)) ? 0.60653067f : r;
  r = ((d == 0) | (d == 4)) ? 0.13533528f : r;
  return r;
}

#if HAVE_TDM
// One TDM descriptor per wave: 2D tile (20x20 f32, stride 1024) -> LDS.
// D# layout per CDNA5 ISA ch.8 (group0: count/lds/global/type,
// group1: data_size/dims/tile/strides).
static __device__ __forceinline__ void tdm_load_tile20(const float* gsrc,
                                                       unsigned lds_addr) {
  const unsigned long long ga = (unsigned long long)(uintptr_t)gsrc;
  u32x4 g0;
  g0[0] = 1u;                                      // count=1, user desc
  g0[1] = lds_addr;                                // LDS byte address
  g0[2] = (unsigned)(ga & 0xFFFFFFFFull);          // global_addr[31:0]
  g0[3] = (unsigned)((ga >> 32) & 0x01FFFFFFull)   // global_addr[56:32]
          | (2u << 30);                            // type=2 ("image")
  i32x8 g1;
  g1[0] = (int)(2u << 16);       // workgroup_mask=0, data_size=2 (4 bytes)
  g1[1] = (int)(1024u << 16);    // tensor_dim0[15:0]=1024
  g1[2] = (int)(1024u << 16);    // tensor_dim0[31:16]=0 | tensor_dim1[15:0]=1024
  g1[3] = (int)(20u << 16);      // tensor_dim1[31:16]=0 | tile_dim0=20
  g1[4] = 20;                    // tile_dim1=20, tile_dim2=0
  g1[5] = 1024;                  // tensor_dim0_stride=1024
  g1[6] = 0;                     // stride0 hi | tensor_dim1_stride lo
  g1[7] = 0;
  const i32x4 z4 = {0, 0, 0, 0};
#if __clang_major__ >= 23
  const i32x8 z8 = {0, 0, 0, 0, 0, 0, 0, 0};
  __builtin_amdgcn_tensor_load_to_lds(g0, g1, z4, z4, z8, 0);
#else
  __builtin_amdgcn_tensor_load_to_lds(g0, g1, z4, z4, 0);
#endif
}
#endif

// ---------------------------------------------------------------------------
// Pass 1: Gaussian blur via WMMA.  One wave per 16x16 output tile.
//   sm = Wv(16x20) * X(20x20) * Wh(20x16)
// Wh[k][n] = g[k-n], Wv[m][k] = g[k-m] (banded, built in registers; the
// per-lane chunk values are identical for both GEMMs -> one table).
// ---------------------------------------------------------------------------
__global__ __launch_bounds__(128) void canny_blur_wmma(
    const float* __restrict__ img, float* __restrict__ sm) {
  __shared__ float XsAll[4][20 * 20];
  __shared__ float TsAll[4][20 * 16];

  const int wave = threadIdx.x >> 5;
  const int lane = threadIdx.x & 31;
  const int tile = blockIdx.x * 4 + wave;           // 65536 tiles total
  const int tx  = tile & 63;
  const int ty  = (tile >> 6) & 63;
  const int b   = tile >> 12;                       // image index 0..15

  const int gx0 = tx * 16, gy0 = ty * 16;
  const float* src = img + (size_t)b * HW;
  float* Xs = XsAll[wave];
  float* Ts = TsAll[wave];

  const int m    = lane & 15;     // row (A) / col (B,C,D) owned by this lane
  const int half = lane >> 4;     // K sub-offset selector

  // Band-weight chunks: wb[kk] = { g(k0-m), g(k0+1-m) }, k0 = 4kk + 2*half.
  // Serves as B (Wh rows @ col m) in GEMM1 and A (Wv row m @ cols) in GEMM2.
  v2f wb[5];
#pragma unroll
  for (int kk = 0; kk < 5; ++kk) {
    const int k0 = kk * 4 + half * 2;
    wb[kk].x = gband(k0 - m);
    wb[kk].y = gband(k0 + 1 - m);
  }

  // ---- Fill the 20x20 input tile (halo 2) in LDS -------------------------
  const bool interior = (tx >= 1) & (tx <= 62) & (ty >= 1) & (ty <= 62);
#if HAVE_TDM
  if (interior) {
    // TDM: one async tensor DMA per wave, EXEC-independent.
    tdm_load_tile20(src + (size_t)(gy0 - 2) * IMG_W + (gx0 - 2),
                    (unsigned)(uintptr_t)Xs);
  } else
#endif
  {
    for (int i = lane; i < 400; i += 32) {
      int r = i / 20, c = i - 20 * (i / 20);
      int gy = gy0 + r - 2, gx = gx0 + c - 2;
      float v = 0.0f;
      if (gy >= 0 && gy < IMG_H && gx >= 0 && gx < IMG_W)
        v = src[gy * IMG_W + gx];
      Xs[i] = v;
    }
  }
#if HAVE_TDM
  __builtin_amdgcn_s_wait_tensorcnt(0);
#endif
  __syncthreads();

  // --- GEMM 1: T(20x16) = X(20x20) * Wh(20x16), two overlapping M=16 groups
  v8f t0 = {};   // rows 0..15 of T
  v8f t1 = {};   // rows 4..19 of T
#pragma unroll
  for (int kk = 0; kk < 5; ++kk) {
    const int k0 = kk * 4 + half * 2;
    v2f a0; a0.x = Xs[m * 20 + k0];       a0.y = Xs[m * 20 + k0 + 1];
    v2f a1; a1.x = Xs[(m + 4) * 20 + k0]; a1.y = Xs[(m + 4) * 20 + k0 + 1];
    t0 = wmma4(a0, wb[kk], t0);
    t1 = wmma4(a1, wb[kk], t1);
  }

  // Spill T to LDS in (row,col) form; overlapping rows write equal values.
#pragma unroll
  for (int v = 0; v < 8; ++v) {
    Ts[(v + 8 * half) * 16 + m]     = t0[v];
    Ts[(4 + v + 8 * half) * 16 + m] = t1[v];
  }
  __syncthreads();

  // --- GEMM 2: S(16x16) = Wv(16x20) * T(20x16)
  v8f s = {};
#pragma unroll
  for (int kk = 0; kk < 5; ++kk) {
    const int k0 = kk * 4 + half * 2;
    v2f bmat; bmat.x = Ts[k0 * 16 + m]; bmat.y = Ts[(k0 + 1) * 16 + m];
    s = wmma4(wb[kk], bmat, s);
  }

  float* dst = sm + (size_t)b * HW + (size_t)gy0 * IMG_W + gx0;
#pragma unroll
  for (int v = 0; v < 8; ++v) {
    dst[(v + 8 * half) * IMG_W + m] = s[v];
  }
}

// ---------------------------------------------------------------------------
// Pass 2: Sobel, gradient magnitude, direction index, per-image max.
// ---------------------------------------------------------------------------
__global__ __launch_bounds__(256) void canny_sobel(
    const float* __restrict__ sm, float* __restrict__ mag,
    unsigned char* __restrict__ dir, unsigned int* __restrict__ permax) {
  const int i = blockIdx.x * 256 + threadIdx.x;
  const int b = i >> 20;
  const int y = (i >> 10) & 1023;
  const int x = i & 1023;
  const float* s = sm + (size_t)b * HW;

  float a[3][3];
#pragma unroll
  for (int dy = -1; dy <= 1; ++dy)
#pragma unroll
    for (int dx = -1; dx <= 1; ++dx) {
      int yy = y + dy, xx = x + dx;
      a[dy + 1][dx + 1] =
          (yy >= 0 && yy < IMG_H && xx >= 0 && xx < IMG_W) ? s[yy * IMG_W + xx]
                                                           : 0.0f;
    }
  const float Ix = (a[0][0] + 2.0f * a[1][0] + a[2][0]) -
                   (a[0][2] + 2.0f * a[1][2] + a[2][2]);
  const float Iy = (a[0][0] + 2.0f * a[0][1] + a[0][2]) -
                   (a[2][0] + 2.0f * a[2][1] + a[2][2]);
  const float m = sqrtf(Ix * Ix + Iy * Iy);

  // direction = round((atan2*180/pi + 180)/45); idx = dir % 8
  const float ang = atan2f(Iy, Ix) * 57.2957764f + 180.0f;
  const int d = (int)rintf(ang * (1.0f / 45.0f));
  const int idx8 = d & 7;

  mag[i] = m;
  dir[i] = (unsigned char)idx8;

  __shared__ unsigned int bm;
  if (threadIdx.x == 0) bm = 0u;
  __syncthreads();
  atomicMax(&bm, __float_as_uint(m));     // m >= 0: bit compare is monotone
  __syncthreads();
  if (threadIdx.x == 0) atomicMax(&permax[b], bm);
}

// ---------------------------------------------------------------------------
// Pass 3: non-max suppression + per-image normalize + global thin max.
// ---------------------------------------------------------------------------
__global__ __launch_bounds__(256) void canny_nms(
    const float* __restrict__ mag, const unsigned char* __restrict__ dir,
    const unsigned int* __restrict__ permax, float* __restrict__ thin,
    unsigned int* __restrict__ thinmax) {
  const int i = blockIdx.x * 256 + threadIdx.x;
  const int b = i >> 20;
  const int y = (i >> 10) & 1023;
  const int x = i & 1023;
  const float* mb = mag + (size_t)b * HW;

  const int offy[8] = {0, 1, 1, 1, 0, -1, -1, -1};
  const int offx[8] = {1, 1, 0, -1, -1, -1, 0, 1};

  const float mc = mb[(y << 10) + x];
  const int k = dir[i];

  auto nbr = [&](int kk) -> float {
    int yy = y + offy[kk], xx = x + offx[kk];
    return (yy >= 0 && yy < IMG_H && xx >= 0 && xx < IMG_W)
               ? mb[yy * IMG_W + xx]
               : 0.0f;
  };
  const float cs_pos = mc - nbr(k);
  const float cs_neg = mc - nbr(k ^ 4);   // (k+4) % 8
  const bool is_max = fminf(cs_pos, cs_neg) > 0.0f;

  const float pm = __uint_as_float(permax[b]);
  const float t = is_max ? (mc / pm) : 0.0f;
  thin[i] = t;

  __shared__ unsigned int bm;
  if (threadIdx.x == 0) bm = 0u;
  __syncthreads();
  atomicMax(&bm, __float_as_uint(t));
  __syncthreads();
  if (threadIdx.x == 0) atomicMax(thinmax, bm);
}

// ---------------------------------------------------------------------------
// Pass 4: double threshold -> uint8 state (0 none, 1 weak, 2 strong).
// ---------------------------------------------------------------------------
__global__ __launch_bounds__(256) void canny_thresh(
    const float* __restrict__ thin, const unsigned int* __restrict__ thinmax,
    unsigned char* __restrict__ state) {
  const int i = blockIdx.x * 256 + threadIdx.x;
  const float hi = __uint_as_float(*thinmax) * HIGH_T;
  const float t = thin[i];
  state[i] = (t >= hi) ? (unsigned char)2
                       : ((t >= LOW_T) ? (unsigned char)1 : (unsigned char)0);
}

// ---------------------------------------------------------------------------
// Pass 5: one hysteresis sweep: 32x32 tile in LDS, iterate to local
// convergence, write back (monotone 1->2 upgrades, in-place is safe).
// ---------------------------------------------------------------------------
__global__ __launch_bounds__(256) void canny_hyst(unsigned char* __restrict__ state) {
  __shared__ unsigned char st[34][34];
  __shared__ int changed;

  const int b = blockIdx.z;
  const int gx0 = blockIdx.x * 32, gy0 = blockIdx.y * 32;
  const int tid = threadIdx.y * 32 + threadIdx.x;
  unsigned char* sb = state + (size_t)b * HW;

  for (int i = tid; i < 34 * 34; i += 256) {
    int r = i / 34, c = i - 34 * (i / 34);
    int gy = gy0 + r - 1, gx = gx0 + c - 1;
    st[r][c] = (gy >= 0 && gy < IMG_H && gx >= 0 && gx < IMG_W)
                   ? sb[gy * IMG_W + gx]
                   : (unsigned char)0;
  }
  __syncthreads();

  const int c = 1 + threadIdx.x;
  for (int it = 0; it < 64; ++it) {
    if (tid == 0) changed = 0;
    __syncthreads();
#pragma unroll
    for (int k = 0; k < 4; ++k) {
      const int r = 1 + threadIdx.y + 8 * k;
      if (st[r][c] == 1) {
        int sn = (st[r - 1][c - 1] >> 1) | (st[r - 1][c] >> 1) |
                 (st[r - 1][c + 1] >> 1) | (st[r][c - 1] >> 1) |
                 (st[r][c + 1] >> 1) | (st[r + 1][c - 1] >> 1) |
                 (st[r + 1][c] >> 1) | (st[r + 1][c + 1] >> 1);
        if (sn & 1) { st[r][c] = 2; changed = 1; }
      }
    }
    __syncthreads();
    const int ch = changed;
    __syncthreads();
    if (!ch) break;
  }

#pragma unroll
  for (int k = 0; k < 4; ++k) {
    const int r = 1 + threadIdx.y + 8 * k;
    sb[(gy0 + r - 1) * IMG_W + (gx0 + c - 1)] = st[r][c];
  }
}

// ---------------------------------------------------------------------------
// Pass 6: finalize: strong -> 255.0, else 0.0
// ---------------------------------------------------------------------------
__global__ __launch_bounds__(256) void canny_final(
    const unsigned char* __restrict__ state, float* __restrict__ out) {
  const int i = blockIdx.x * 256 + threadIdx.x;
  out[i] = (state[i] == 2) ? 255.0f : 0.0f;
}

__global__ void canny_init(unsigned int* __restrict__ atom) {
  if (threadIdx.x < 32) atom[threadIdx.x] = 0u;
}

// ---------------------------------------------------------------------------
extern "C" void kernel_launch(void* const* d_in, const int* in_sizes, int n_in,
                              void* d_out, int out_size, void* d_ws, size_t ws_size,
                              hipStream_t stream) {
  (void)in_sizes; (void)n_in; (void)out_size; (void)ws_size;
  const float* img = (const float*)d_in[0];
  float* out = (float*)d_out;

  char* ws = (char*)d_ws;
  float*         sm    = (float*)ws;                         // 64 MB
  float*         mag   = (float*)(ws + NPIX * 4);            // 64 MB
  unsigned char* st8   = (unsigned char*)(ws + NPIX * 8);    // 16 MB (dir, then state)
  unsigned int*  atom  = (unsigned int*)(ws + NPIX * 9);     // [0..15] permax, [16] thinmax
  float*         thin  = sm;                                 // reuse: sm dead after sobel

  const int nb = (int)(NPIX / 256);   // 65536 blocks for pointwise passes

  canny_init<<<1, 32, 0, stream>>>(atom);
  canny_blur_wmma<<<65536 / 4, 128, 0, stream>>>(img, sm);
  canny_sobel<<<nb, 256, 0, stream>>>(sm, mag, st8, atom);
  canny_nms<<<nb, 256, 0, stream>>>(mag, st8, atom, thin, atom + 16);
  canny_thresh<<<nb, 256, 0, stream>>>(thin, atom + 16, st8);
  dim3 hgrid(IMG_W / 32, IMG_H / 32, BATCH);
  dim3 hblk(32, 8);
  for (int it = 0; it < HYST_ITERS; ++it)
    canny_hyst<<<hgrid, hblk, 0, stream>>>(st8);
  canny_final<<<nb, 256, 0, stream>>>(st8, out);
}